// SpatialModel_6116033429578
// MI455X (gfx1250) — compile-verified
//
#include <hip/hip_runtime.h>
#include <math.h>

typedef __attribute__((ext_vector_type(2))) float v2f;
typedef __attribute__((ext_vector_type(8))) float v8f;

// ---- ordered-uint encoding for float atomic max ----
__device__ __forceinline__ unsigned f2ord(float f) {
    unsigned u = __float_as_uint(f);
    return (u & 0x80000000u) ? ~u : (u | 0x80000000u);
}
__device__ __forceinline__ float ord2f(unsigned u) {
    unsigned v = (u & 0x80000000u) ? (u & 0x7FFFFFFFu) : ~u;
    return __uint_as_float(v);
}

// ---- dense GEMM H = X(N x FIN) * W(FIN x FOUT), f32 WMMA 16x16x4 ----
// Static shapes; W staged in LDS (pad rows FIN..FINP-1 zeroed so B needs no
// guards); one wave computes a full 16 x FOUT strip (A fragment reused across
// all column tiles). Full-tile waves take a branch-free store path.
template <int FIN, int FINP, int FOUT>
__global__ void gemm_wmma(const float* __restrict__ X, const float* __restrict__ W,
                          float* __restrict__ H, int N) {
    constexpr int KT = FINP / 4;   // K steps of 4
    constexpr int NT = FOUT / 16;  // column tiles per wave
    __shared__ float Ws[FINP * FOUT];
    for (int i = threadIdx.x; i < FIN * FOUT; i += blockDim.x) Ws[i] = W[i];
    if (FINP > FIN)
        for (int i = FIN * FOUT + threadIdx.x; i < FINP * FOUT; i += blockDim.x)
            Ws[i] = 0.f;
    __syncthreads();

    int lane = threadIdx.x & 31;
    int wid  = blockIdx.x * (blockDim.x >> 5) + (threadIdx.x >> 5);
    int rowTiles = (N + 15) >> 4;
    if (wid >= rowTiles) return;               // wave-uniform: EXEC stays all-ones

    int colL = lane & 15;                      // B/C: lane%16 = column within tile
    int row  = (wid << 4) + colL;              // A: lane%16 = row (M)
    int rowc = row < N ? row : N - 1;          // clamp for safe loads
    int koff = (lane >> 4) << 1;               // lanes 16-31 hold K+2, K+3

    v8f acc[NT];
#pragma unroll
    for (int t = 0; t < NT; ++t) acc[t] = v8f{0.f,0.f,0.f,0.f,0.f,0.f,0.f,0.f};

    const float* Xr = X + (long)rowc * FIN;
#pragma unroll
    for (int kt = 0; kt < KT; ++kt) {
        int k0 = kt * 4 + koff;
        int k1 = k0 + 1;
        v2f a;
        if constexpr (FIN % 4 == 0) {
            a = *(const v2f*)(Xr + k0);        // contiguous, 8B aligned -> b64
        } else {                               // FIN=7 tail: clamp + select
            float v0 = Xr[k0 < FIN ? k0 : 0];
            float v1 = Xr[k1 < FIN ? k1 : 0];
            a.x = (k0 < FIN) ? v0 : 0.f;
            a.y = (k1 < FIN) ? v1 : 0.f;
        }
#pragma unroll
        for (int t = 0; t < NT; ++t) {
            v2f b;
            b.x = Ws[k0 * FOUT + t * 16 + colL];
            b.y = Ws[k1 * FOUT + t * 16 + colL];
            acc[t] = __builtin_amdgcn_wmma_f32_16x16x4_f32(false, a, false, b,
                                                           (short)0, acc[t],
                                                           false, false);
        }
    }
    // C/D layout: VGPR v -> M = v (lanes 0-15) / v+8 (lanes 16-31), N = lane%16
    int rbase = (wid << 4) + ((lane >> 4) << 3);
    if (((wid + 1) << 4) <= N) {               // wave-uniform fast path: no guards
        float* Hp = H + (long)rbase * FOUT + colL;
#pragma unroll
        for (int t = 0; t < NT; ++t)
#pragma unroll
            for (int v = 0; v < 8; ++v)
                Hp[(long)v * FOUT + t * 16] = acc[t][v];
    } else {                                   // tail tile (unused when N%16==0)
#pragma unroll
        for (int t = 0; t < NT; ++t)
#pragma unroll
            for (int v = 0; v < 8; ++v) {
                int r = rbase + v;
                if (r < N) H[(long)r * FOUT + t * 16 + colL] = acc[t][v];
            }
    }
}

// ---- degree + sum of edge attrs per dst ----
__global__ void deg_loop_accum(const int* __restrict__ dst, const float* __restrict__ ecol,
                               float* __restrict__ deg, float* __restrict__ lsum, int E) {
    int e = blockIdx.x * blockDim.x + threadIdx.x;
    if (e >= E) return;
    int d = dst[e];
    atomicAdd(&deg[d], 1.0f);
    atomicAdd(&lsum[d], ecol[e]);
}

__global__ void loop_div(float* __restrict__ lsum, const float* __restrict__ deg, int N) {
    int i = blockIdx.x * blockDim.x + threadIdx.x;
    if (i < N) lsum[i] = lsum[i] / fmaxf(deg[i], 1.0f);
}

// ---- per-node scores hs = h.a_src, hd = h.a_dst (one wave per node) ----
__global__ void node_scores(const float* __restrict__ H, const float* __restrict__ a_src,
                            const float* __restrict__ a_dst, float* __restrict__ hs,
                            float* __restrict__ hd, int N, int fout) {
    int wid  = blockIdx.x * (blockDim.x >> 5) + (threadIdx.x >> 5);
    int lane = threadIdx.x & 31;
    if (wid >= N) return;
    float s = 0.f, d = 0.f;
    for (int j = lane; j < fout; j += 32) {
        float v = H[(long)wid * fout + j];
        s += v * a_src[j];
        d += v * a_dst[j];
    }
    for (int o = 16; o > 0; o >>= 1) {
        s += __shfl_down(s, o, 32);
        d += __shfl_down(d, o, 32);
    }
    if (lane == 0) { hs[wid] = s; hd[wid] = d; }
}

// ---- scalar dot (edge_dim==1 so (ea@We)@a_e == ea * dot(We, a_e)) ----
__global__ void dot_small(const float* __restrict__ a, const float* __restrict__ b,
                          float* __restrict__ out, int n) {
    int lane = threadIdx.x;
    float s = 0.f;
    for (int j = lane; j < n; j += 32) s += a[j] * b[j];
    for (int o = 16; o > 0; o >>= 1) s += __shfl_down(s, o, 32);
    if (lane == 0) out[0] = s;
}

// ---- alpha = leaky_relu(hs[s] + hd[d] + ea*cWA), segment max via ordered atomic ----
__global__ void alpha_kernel(const int* __restrict__ src, const int* __restrict__ dst,
                             const float* __restrict__ ecol, const float* __restrict__ lattr,
                             const float* __restrict__ hs, const float* __restrict__ hd,
                             const float* __restrict__ cWA, float* __restrict__ alpha,
                             unsigned* __restrict__ ordmax, int E, int N) {
    int i = blockIdx.x * blockDim.x + threadIdx.x;
    if (i >= E + N) return;
    float c = cWA[0];
    int d; float a;
    if (i < E) {
        d = dst[i];
        a = hs[src[i]] + hd[d] + ecol[i] * c;
    } else {
        d = i - E;
        a = hs[d] + hd[d] + lattr[d] * c;
    }
    a = (a > 0.f) ? a : 0.2f * a;
    alpha[i] = a;
    atomicMax(&ordmax[d], f2ord(a));
}

// ---- ex = exp(alpha - amax[d]); denom[d] += ex ----
__global__ void exp_kernel(const int* __restrict__ dst, const float* __restrict__ alpha,
                           const unsigned* __restrict__ ordmax, float* __restrict__ ex,
                           float* __restrict__ denom, int E, int N) {
    int i = blockIdx.x * blockDim.x + threadIdx.x;
    if (i >= E + N) return;
    int d = (i < E) ? dst[i] : (i - E);
    float e = expf(alpha[i] - ord2f(ordmax[d]));
    ex[i] = e;
    atomicAdd(&denom[d], e);
}

// ---- out[n,:] = w_self[n] * h[n,:]  (non-atomic accumulator init) ----
template <int FOUT>
__global__ void init_self(const float* __restrict__ H, const float* __restrict__ ex,
                          const float* __restrict__ denom, float* __restrict__ out,
                          int N, int E) {
    int i = blockIdx.x * blockDim.x + threadIdx.x;
    if (i >= N * FOUT) return;
    int n = i / FOUT;
    out[i] = (ex[E + n] / denom[n]) * H[i];
}

// ---- out[d,:] += w[e] * h[s,:]  (wave per edge, FOUT/32 contiguous per lane) ----
template <int FOUT>
__global__ void aggr_edges(const int* __restrict__ src, const int* __restrict__ dst,
                           const float* __restrict__ ex, const float* __restrict__ denom,
                           const float* __restrict__ H, float* __restrict__ out, int E) {
    constexpr int VPL = FOUT / 32;  // 4 (b128 gather) or 2 (b64 gather)
    int wid  = blockIdx.x * (blockDim.x >> 5) + (threadIdx.x >> 5);
    int lane = threadIdx.x & 31;
    if (wid >= E) return;
    int s = src[wid], d = dst[wid];
    float w = ex[wid] / denom[d];
    const float* hrow = H + (long)s * FOUT + lane * VPL;
    float* orow = out + (long)d * FOUT + lane * VPL;
    float v[VPL];
#pragma unroll
    for (int j = 0; j < VPL; ++j) v[j] = hrow[j];
#pragma unroll
    for (int j = 0; j < VPL; ++j) atomicAdd(orow + j, w * v[j]);
}

// ---- x = elu(aggr + b) in place ----
template <int FOUT>
__global__ void finalize_elu(float* __restrict__ p, const float* __restrict__ b, int N) {
    int i = blockIdx.x * blockDim.x + threadIdx.x;
    if (i >= N * FOUT) return;
    float v = p[i] + b[i & (FOUT - 1)];
    p[i] = (v > 0.f) ? v : (expf(v) - 1.f);
}

// ---- column mean, fout=64: coalesced partial sums + per-column atomics ----
// meansum must be zeroed beforehand; each block adds its partial * (1/N).
__global__ void colmean_acc(const float* __restrict__ X, float* __restrict__ meansum,
                            int N, float invN) {
    __shared__ float red[256];
    int c    = threadIdx.x & 63;               // column
    int rsub = threadIdx.x >> 6;               // 0..3: row within group of 4
    float s = 0.f;
    for (long r = (long)blockIdx.x * 4 + rsub; r < N; r += (long)gridDim.x * 4)
        s += X[r * 64 + c];                    // fully coalesced
    red[threadIdx.x] = s;
    __syncthreads();
    if (threadIdx.x < 128) red[threadIdx.x] += red[threadIdx.x + 128];
    __syncthreads();
    if (threadIdx.x < 64)
        atomicAdd(&meansum[c], (red[threadIdx.x] + red[threadIdx.x + 64]) * invN);
}

// ---- head MLP: concat->linear->concat->relu linear->linear (single block) ----
__global__ void head_mlp(const float* __restrict__ m1, const float* __restrict__ m2,
                         const float* __restrict__ xn1, const float* __restrict__ xn2,
                         const float* __restrict__ W_lin, const float* __restrict__ b_lin,
                         const float* __restrict__ W_c1, const float* __restrict__ b_c1,
                         const float* __restrict__ W_c2, const float* __restrict__ b_c2,
                         float* __restrict__ out) {
    __shared__ float in1[80], in2[80], g[128], h[16];
    int t = threadIdx.x;
    if (t < 64)      { in1[t] = m1[t];        in2[t] = m2[t]; }
    else if (t < 80) { in1[t] = xn1[t - 64];  in2[t] = xn2[t - 64]; }
    __syncthreads();
    if (t < 64) {
        float s1 = b_lin[t], s2 = b_lin[t];
        for (int k = 0; k < 80; ++k) {
            float w = W_lin[k * 64 + t];
            s1 += in1[k] * w;
            s2 += in2[k] * w;
        }
        g[t] = s1; g[64 + t] = s2;
    }
    __syncthreads();
    if (t < 16) {
        float s = b_c1[t];
        for (int k = 0; k < 128; ++k) s += g[k] * W_c1[k * 16 + t];
        h[t] = fmaxf(s, 0.f);
    }
    __syncthreads();
    if (t < 3) {
        float s = b_c2[t];
        for (int k = 0; k < 16; ++k) s += h[k] * W_c2[k * 3 + t];
        out[t] = s;
    }
}

extern "C" void kernel_launch(void* const* d_in, const int* in_sizes, int n_in,
                              void* d_out, int out_size, void* d_ws, size_t ws_size,
                              hipStream_t stream) {
    const float* x1  = (const float*)d_in[0];
    const float* x2  = (const float*)d_in[1];
    const int*   ei1 = (const int*)d_in[2];
    const int*   ei2 = (const int*)d_in[3];
    const float* xn1 = (const float*)d_in[4];
    const float* xn2 = (const float*)d_in[5];
    const float* ec1 = (const float*)d_in[6];
    const float* ec2 = (const float*)d_in[7];
    const float *W[3], *asrc[3], *adst[3], *We[3], *ae[3], *bb[3];
    for (int l = 0; l < 3; ++l) {
        W[l]    = (const float*)d_in[8 + 6 * l + 0];
        asrc[l] = (const float*)d_in[8 + 6 * l + 1];
        adst[l] = (const float*)d_in[8 + 6 * l + 2];
        We[l]   = (const float*)d_in[8 + 6 * l + 3];
        ae[l]   = (const float*)d_in[8 + 6 * l + 4];
        bb[l]   = (const float*)d_in[8 + 6 * l + 5];
    }
    const float* W_lin = (const float*)d_in[26];
    const float* b_lin = (const float*)d_in[27];
    const float* W_c1  = (const float*)d_in[28];
    const float* b_c1  = (const float*)d_in[29];
    const float* W_c2  = (const float*)d_in[30];
    const float* b_c2  = (const float*)d_in[31];

    const int N = in_sizes[0] / 7;
    const int E = in_sizes[2] / 2;

    // carve workspace
    char* ws = (char*)d_ws;
    size_t off = 0;
    auto carve = [&](size_t bytes) -> void* {
        void* p = ws + off;
        off = (off + bytes + 255) & ~(size_t)255;
        return p;
    };
    float*    bufA   = (float*)carve((size_t)N * 128 * sizeof(float));
    float*    bufB   = (float*)carve((size_t)N * 128 * sizeof(float));
    float*    hs     = (float*)carve((size_t)N * sizeof(float));
    float*    hd     = (float*)carve((size_t)N * sizeof(float));
    float*    deg    = (float*)carve((size_t)N * sizeof(float));
    float*    lattr  = (float*)carve((size_t)N * sizeof(float));
    unsigned* ordmax = (unsigned*)carve((size_t)N * sizeof(unsigned));
    float*    denom  = (float*)carve((size_t)N * sizeof(float));
    float*    alpha  = (float*)carve((size_t)(E + N) * sizeof(float));
    float*    ex     = (float*)carve((size_t)(E + N) * sizeof(float));
    float*    cWA    = (float*)carve(sizeof(float));
    float*    mean1  = (float*)carve(64 * sizeof(float));
    float*    mean2  = (float*)carve(64 * sizeof(float));

    const int tb  = 256;
    const int wpb = tb / 32;
    const int rowBlocks = (((N + 15) / 16) + wpb - 1) / wpb;

    auto run_graph = [&](const float* x0, const int* ei, const float* ecol, float* meanOut) {
        const int* src = ei;
        const int* dst = ei + E;
        hipMemsetAsync(deg,   0, (size_t)N * sizeof(float), stream);
        hipMemsetAsync(lattr, 0, (size_t)N * sizeof(float), stream);
        deg_loop_accum<<<(E + tb - 1) / tb, tb, 0, stream>>>(dst, ecol, deg, lattr, E);
        loop_div<<<(N + tb - 1) / tb, tb, 0, stream>>>(lattr, deg, N);

        const float* x = x0;
        for (int l = 0; l < 3; ++l) {
            const int fout = (l == 2) ? 64 : 128;
            if (l == 0)
                gemm_wmma<7, 8, 128><<<rowBlocks, tb, 0, stream>>>(x, W[l], bufB, N);
            else if (l == 1)
                gemm_wmma<128, 128, 128><<<rowBlocks, tb, 0, stream>>>(x, W[l], bufB, N);
            else
                gemm_wmma<128, 128, 64><<<rowBlocks, tb, 0, stream>>>(x, W[l], bufB, N);
            node_scores<<<(N + wpb - 1) / wpb, tb, 0, stream>>>(bufB, asrc[l], adst[l],
                                                                hs, hd, N, fout);
            dot_small<<<1, 32, 0, stream>>>(We[l], ae[l], cWA, fout);
            hipMemsetAsync(ordmax, 0, (size_t)N * sizeof(unsigned), stream);
            hipMemsetAsync(denom,  0, (size_t)N * sizeof(float), stream);
            int T = E + N;
            alpha_kernel<<<(T + tb - 1) / tb, tb, 0, stream>>>(src, dst, ecol, lattr, hs, hd,
                                                               cWA, alpha, ordmax, E, N);
            exp_kernel<<<(T + tb - 1) / tb, tb, 0, stream>>>(dst, alpha, ordmax, ex, denom,
                                                             E, N);
            long NF = (long)N * fout;
            if (fout == 128) {
                init_self<128><<<(NF + tb - 1) / tb, tb, 0, stream>>>(bufB, ex, denom, bufA,
                                                                      N, E);
                aggr_edges<128><<<(E + wpb - 1) / wpb, tb, 0, stream>>>(src, dst, ex, denom,
                                                                        bufB, bufA, E);
                finalize_elu<128><<<(NF + tb - 1) / tb, tb, 0, stream>>>(bufA, bb[l], N);
            } else {
                init_self<64><<<(NF + tb - 1) / tb, tb, 0, stream>>>(bufB, ex, denom, bufA,
                                                                     N, E);
                aggr_edges<64><<<(E + wpb - 1) / wpb, tb, 0, stream>>>(src, dst, ex, denom,
                                                                       bufB, bufA, E);
                finalize_elu<64><<<(NF + tb - 1) / tb, tb, 0, stream>>>(bufA, bb[l], N);
            }
            x = bufA;
        }
        hipMemsetAsync(meanOut, 0, 64 * sizeof(float), stream);
        colmean_acc<<<256, 256, 0, stream>>>(bufA, meanOut, N, 1.0f / (float)N);
    };

    run_graph(x1, ei1, ec1, mean1);
    run_graph(x2, ei2, ec2, mean2);
    head_mlp<<<1, 128, 0, stream>>>(mean1, mean2, xn1, xn2, W_lin, b_lin,
                                    W_c1, b_c1, W_c2, b_c2, (float*)d_out);
}